// CustomGCN_64733747085460
// MI455X (gfx1250) — compile-verified
//
#include <hip/hip_runtime.h>
#include <stdint.h>

typedef float v2f __attribute__((ext_vector_type(2)));
typedef float v8f __attribute__((ext_vector_type(8)));

// ---------------- degree / normalization precompute ----------------
__global__ void k_deg_init(float* __restrict__ deg, int n) {
  int i = blockIdx.x * blockDim.x + threadIdx.x;
  if (i < n) deg[i] = 1.0f;  // self-loop weight contributes 1 to every target degree
}

__global__ void k_deg_accum(const long long* __restrict__ col, const float* __restrict__ ew,
                            float* __restrict__ deg, int E) {
  int e = blockIdx.x * blockDim.x + threadIdx.x;
  if (e < E) atomicAdd(&deg[col[e]], ew[e]);
}

__global__ void k_dinv(float* __restrict__ deg, int n) {
  int i = blockIdx.x * blockDim.x + threadIdx.x;
  if (i < n) { float d = deg[i]; deg[i] = (d > 0.0f) ? rsqrtf(d) : 0.0f; }
}

__global__ void k_norm(const long long* __restrict__ row, const long long* __restrict__ col,
                       const float* __restrict__ ew, const float* __restrict__ dinv,
                       float* __restrict__ norm, int E) {
  int e = blockIdx.x * blockDim.x + threadIdx.x;
  if (e < E) norm[e] = dinv[row[e]] * ew[e] * dinv[col[e]];
}

// ---------------- fp32 WMMA GEMM: XW[n,OUTD] = A[n,K] @ W[K,OUTD] ----------------
// One wave computes one 16x16 output tile via V_WMMA_F32_16X16X4_F32.
// K and OUTD are compile-time -> fully unrolled K loop, immediate-offset loads,
// no per-iteration address math, no EXEC divergence (loads clamped, not predicated).
// A (16x4 f32): lane&15 = M row; lanes<16 hold K=k0..k0+1, lanes>=16 hold K=k0+2..k0+3.
// B (4x16):     lane&15 = N col; same K split across wave halves.
// C/D (16x16):  lane&15 = N col; VGPR v = row v (lanes 0-15) or v+8 (lanes 16-31).
template <int K, int OUTD>
__global__ void k_gemm_wmma_f32(const float* __restrict__ A, const float* __restrict__ W,
                                float* __restrict__ XW, int n) {
  const int lane  = threadIdx.x & 31;
  const int wave  = threadIdx.x >> 5;
  const int mtile = blockIdx.x * 8 + wave;
  const int ntile = blockIdx.y;
  if (mtile * 16 >= n) return;  // wave-uniform early out
  const int l15  = lane & 15;
  const int half = lane >> 4;
  int arow = mtile * 16 + l15;
  arow = (arow < n) ? arow : (n - 1);  // clamp: keeps loads in-bounds, no EXEC churn
  const int bcol = ntile * 16 + l15;

  const float* __restrict__ ap = A + (size_t)arow * K + half * 2;
  const float* __restrict__ bp = W + (size_t)(half * 2) * OUTD + bcol;

  v8f c = {};
#pragma unroll
  for (int k0 = 0; k0 < K; k0 += 4) {
    v2f a = *reinterpret_cast<const v2f*>(ap + k0);  // contiguous pair -> b64 load
    v2f b;
    b.x = bp[(size_t)k0 * OUTD];
    b.y = bp[(size_t)k0 * OUTD + OUTD];
    c = __builtin_amdgcn_wmma_f32_16x16x4_f32(
        /*neg_a=*/false, a, /*neg_b=*/false, b,
        /*c_mod=*/(short)0, c, /*reuse_a=*/false, /*reuse_b=*/false);
  }

  const int ccol = ntile * 16 + l15;
  if ((mtile + 1) * 16 <= n) {  // full tile (always true for n % 16 == 0)
    float* __restrict__ outp = XW + (size_t)(mtile * 16 + half * 8) * OUTD + ccol;
#pragma unroll
    for (int v = 0; v < 8; ++v) outp[(size_t)v * OUTD] = c[v];
  } else {
#pragma unroll
    for (int v = 0; v < 8; ++v) {
      int m = mtile * 16 + v + half * 8;
      if (m < n) XW[(size_t)m * OUTD + ccol] = c[v];
    }
  }
}

// ---------------- out = bias + dinv^2 * xw (self-loop term; full init of out) ----------------
__global__ void k_init_out(const float* __restrict__ xw, const float* __restrict__ dinv,
                           const float* __restrict__ bias, float* __restrict__ out,
                           int total, int od_shift) {
  int tid = blockIdx.x * blockDim.x + threadIdx.x;
  if (tid >= total) return;
  int i = tid >> od_shift;
  int j = tid & ((1 << od_shift) - 1);
  float di = dinv[i];
  out[tid] = bias[j] + di * di * xw[tid];
}

// ---------------- edge scatter: out[col] += norm[e] * xw[row] ----------------
// One thread = one edge x 4 contiguous features (float4 gather + 4x global f32 atomics).
__global__ void k_scatter(const long long* __restrict__ row, const long long* __restrict__ col,
                          const float* __restrict__ norm, const float* __restrict__ xw,
                          float* __restrict__ out, int total, int od_shift) {
  const int cshift = od_shift - 2;  // chunks of 4 floats per edge
  int tid = blockIdx.x * blockDim.x + threadIdx.x;
  if (tid >= total) return;
  int e  = tid >> cshift;
  int j4 = (tid & ((1 << cshift) - 1)) << 2;
  long long r  = row[e];
  long long cc = col[e];
  float nv = norm[e];
  const float4 v = *reinterpret_cast<const float4*>(xw + ((size_t)r << od_shift) + j4);
  float* dst = out + ((size_t)cc << od_shift) + j4;
  atomicAdd(dst + 0, nv * v.x);
  atomicAdd(dst + 1, nv * v.y);
  atomicAdd(dst + 2, nv * v.z);
  atomicAdd(dst + 3, nv * v.w);
}

__global__ void k_tanh(const float* __restrict__ in, float* __restrict__ out, int cnt) {
  int i = blockIdx.x * blockDim.x + threadIdx.x;
  if (i < cnt) out[i] = tanhf(in[i]);
}

// ---------------- launcher ----------------
extern "C" void kernel_launch(void* const* d_in, const int* in_sizes, int n_in,
                              void* d_out, int out_size, void* d_ws, size_t ws_size,
                              hipStream_t stream) {
  const float*     x  = (const float*)d_in[0];      // [n,128] f32
  const long long* ei = (const long long*)d_in[1];  // [2,E] int64
  const float*     ew = (const float*)d_in[2];      // [E] f32
  const float*     W1 = (const float*)d_in[3];      // [128,64]
  const float*     b1 = (const float*)d_in[4];
  const float*     W2 = (const float*)d_in[5];      // [64,32]
  const float*     b2 = (const float*)d_in[6];
  const float*     W3 = (const float*)d_in[7];      // [32,16]
  const float*     b3 = (const float*)d_in[8];

  const int n = in_sizes[0] / 128;
  const int E = in_sizes[1] / 2;
  const long long* row = ei;      // edge_index[0]
  const long long* col = ei + E;  // edge_index[1]

  float* out    = (float*)d_out;
  float* logits = out;                   // [n,16]
  float* emb    = out + (size_t)n * 16;  // [n,32]

  // workspace layout (floats): dinv[n] | norm[E] | xw[n*64] | h[n*64]
  float* ws   = (float*)d_ws;
  float* dinv = ws;
  float* norm = dinv + n;
  float* xw   = norm + E;
  float* hbuf = xw + (size_t)n * 64;

  const int T = 256;
  auto cdiv = [](long long a, long long b) { return (int)((a + b - 1) / b); };

  // degree / dinv / per-edge norm (identical across all three layers; compute once)
  k_deg_init <<<cdiv(n, T), T, 0, stream>>>(dinv, n);
  k_deg_accum<<<cdiv(E, T), T, 0, stream>>>(col, ew, dinv, E);
  k_dinv     <<<cdiv(n, T), T, 0, stream>>>(dinv, n);
  k_norm     <<<cdiv(E, T), T, 0, stream>>>(row, col, ew, dinv, norm, E);

  const int mblocks = cdiv((n + 15) / 16, 8);

  // ---- layer 1: [n,128] @ [128,64], aggregate, tanh -> hbuf ----
  {
    dim3 g(mblocks, 64 / 16);
    k_gemm_wmma_f32<128, 64><<<g, 256, 0, stream>>>(x, W1, xw, n);
    k_init_out<<<cdiv((long long)n * 64, T), T, 0, stream>>>(xw, dinv, b1, hbuf, n * 64, 6);
    k_scatter <<<cdiv((long long)E * 16, T), T, 0, stream>>>(row, col, norm, xw, hbuf, E * 16, 6);
    k_tanh    <<<cdiv((long long)n * 64, T), T, 0, stream>>>(hbuf, hbuf, n * 64);
  }
  // ---- layer 2: [n,64] @ [64,32], aggregate -> emb (output); h_act = tanh(emb) -> hbuf ----
  {
    dim3 g(mblocks, 32 / 16);
    k_gemm_wmma_f32<64, 32><<<g, 256, 0, stream>>>(hbuf, W2, xw, n);
    k_init_out<<<cdiv((long long)n * 32, T), T, 0, stream>>>(xw, dinv, b2, emb, n * 32, 5);
    k_scatter <<<cdiv((long long)E * 8, T), T, 0, stream>>>(row, col, norm, xw, emb, E * 8, 5);
    k_tanh    <<<cdiv((long long)n * 32, T), T, 0, stream>>>(emb, hbuf, n * 32);
  }
  // ---- layer 3: [n,32] @ [32,16], aggregate -> logits (output) ----
  {
    dim3 g(mblocks, 16 / 16);
    k_gemm_wmma_f32<32, 16><<<g, 256, 0, stream>>>(hbuf, W3, xw, n);
    k_init_out<<<cdiv((long long)n * 16, T), T, 0, stream>>>(xw, dinv, b3, logits, n * 16, 4);
    k_scatter <<<cdiv((long long)E * 4, T), T, 0, stream>>>(row, col, norm, xw, logits, E * 4, 4);
  }
}